// BlockUpdateBlock_62148176773669
// MI455X (gfx1250) — compile-verified
//
#include <hip/hip_runtime.h>
#include <hip/hip_bf16.h>

// ---------------------------------------------------------------------------
// e3nn 'uvu' tensor product, CDNA5 (gfx1250), f32 WMMA 16x16x4 implementation.
//   block : [n,480]  (128x0e + 64x1o + 32x2e)
//   others: [n,240]  (64x0e + 32x1e + 16x2e)
//   out   : [n,480] new_block  ++  [n,240] others (copy)
// Design: per-wave persistent WMMA A-fragments (weights), others staged
// transposed into LDS via async global->LDS DMA, one wave owns all paths of
// an (output irrep, 16-u tile); GEMMs sharing a B stream are fused so one
// ds_load feeds up to 3 WMMAs; B loads software-pipelined 2 deep.
// Wigner-3j tensors generated at compile time (constexpr Racah formula).
// ---------------------------------------------------------------------------

#define DB 480
#define DO 240
#define CHUNK_ROWS 32
#define TILE_WORDS (DO * 16)     // one 16-row tile: yt[c][r], pitch 16 words

typedef float v2f __attribute__((ext_vector_type(2)));
typedef float v8f __attribute__((ext_vector_type(8)));

#if __has_builtin(__builtin_amdgcn_sched_barrier)
#define SCHED_FENCE() __builtin_amdgcn_sched_barrier(0)
#else
#define SCHED_FENCE()
#endif

#define WMMA_F32X4(A, B, D) \
  __builtin_amdgcn_wmma_f32_16x16x4_f32(false, (A), false, (B), (short)0, (D), false, false)

// ======================= compile-time real Wigner 3j ========================
struct CD { double re, im; };
constexpr CD cmul(CD a, CD b) { return {a.re*b.re - a.im*b.im, a.re*b.im + a.im*b.re}; }
constexpr double cabs_(double x) { return x < 0 ? -x : x; }
constexpr double cfact(int n) { double r = 1.0; for (int i = 2; i <= n; ++i) r *= (double)i; return r; }
constexpr double csqrt_(double x) {
  if (x <= 0.0) return 0.0;
  double g = x > 1.0 ? x : 1.0;
  for (int i = 0; i < 100; ++i) g = 0.5 * (g + x / g);
  return g;
}
constexpr double neg1p(int n) { return (n % 2 == 0) ? 1.0 : -1.0; }

constexpr double w3j_c(int j1, int j2, int j3, int m1, int m2, int m3) {
  if (m1 + m2 + m3 != 0) return 0.0;
  const int lo = (j1 > j2) ? (j1 - j2) : (j2 - j1);
  if (j3 < lo || j3 > j1 + j2) return 0.0;
  if (m1 < -j1 || m1 > j1 || m2 < -j2 || m2 > j2 || m3 < -j3 || m3 > j3) return 0.0;
  const double pre = neg1p(j1 - j2 - m3) * csqrt_(
      cfact(j1 + j2 - j3) * cfact(j1 - j2 + j3) * cfact(-j1 + j2 + j3) / cfact(j1 + j2 + j3 + 1)
      * cfact(j1 + m1) * cfact(j1 - m1) * cfact(j2 + m2) * cfact(j2 - m2)
      * cfact(j3 + m3) * cfact(j3 - m3));
  int kmin = 0;
  if (j2 - j3 - m1 > kmin) kmin = j2 - j3 - m1;
  if (j1 - j3 + m2 > kmin) kmin = j1 - j3 + m2;
  int kmax = j1 + j2 - j3;
  if (j1 - m1 < kmax) kmax = j1 - m1;
  if (j2 + m2 < kmax) kmax = j2 + m2;
  double s = 0.0;
  for (int k = kmin; k <= kmax; ++k)
    s += neg1p(k) / (cfact(k) * cfact(j1 + j2 - j3 - k) * cfact(j1 - m1 - k)
                     * cfact(j2 + m2 - k) * cfact(j3 - j2 + m1 + k) * cfact(j3 - j1 - m2 + k));
  return pre * s;
}

// complex -> real spherical harmonics change of basis, row r = m+l
constexpr CD Uent(int l, int r, int c) {
  const int m = r - l;
  const double s2i = 1.0 / csqrt_(2.0);
  CD z{0.0, 0.0};
  if (m == 0) { if (c == l) z = {1.0, 0.0}; }
  else if (m > 0) {
    if (c == l - m) z.re += s2i;
    if (c == l + m) z.re += neg1p(m) * s2i;
  } else {
    const int a = -m;
    if (c == l - a) z.im += s2i;
    if (c == l + a) z.im += -neg1p(a) * s2i;
  }
  return z;
}

template<int L1, int L2, int L3> struct W3J { float c[2*L1+1][2*L2+1][2*L3+1]; };

template<int L1, int L2, int L3>
constexpr W3J<L1, L2, L3> make_w3j(double scale) {
  constexpr int D1 = 2*L1+1, D2 = 2*L2+1, D3 = 2*L3+1;
  double Tre[D1][D2][D3] = {}, Tim[D1][D2][D3] = {};
  for (int i = 0; i < D1; ++i)
    for (int j = 0; j < D2; ++j)
      for (int k = 0; k < D3; ++k) {
        CD acc{0.0, 0.0};
        for (int a = 0; a < D1; ++a) {
          const CD u1 = Uent(L1, i, a); if (u1.re == 0.0 && u1.im == 0.0) continue;
          for (int b = 0; b < D2; ++b) {
            const CD u2 = Uent(L2, j, b); if (u2.re == 0.0 && u2.im == 0.0) continue;
            const CD u12 = cmul(u1, u2);
            for (int c2 = 0; c2 < D3; ++c2) {
              const CD u3 = Uent(L3, k, c2); if (u3.re == 0.0 && u3.im == 0.0) continue;
              const double w = w3j_c(L1, L2, L3, a - L1, b - L2, c2 - L3);
              if (w == 0.0) continue;
              const CD t = cmul(u12, u3);
              acc.re += t.re * w; acc.im += t.im * w;
            }
          }
        }
        Tre[i][j][k] = acc.re; Tim[i][j][k] = acc.im;
      }
  double mre = 0.0, mim = 0.0;
  for (int i = 0; i < D1; ++i) for (int j = 0; j < D2; ++j) for (int k = 0; k < D3; ++k) {
    if (cabs_(Tre[i][j][k]) > mre) mre = cabs_(Tre[i][j][k]);
    if (cabs_(Tim[i][j][k]) > mim) mim = cabs_(Tim[i][j][k]);
  }
  const bool useRe = (mre >= mim);
  double nrm = 0.0;
  for (int i = 0; i < D1; ++i) for (int j = 0; j < D2; ++j) for (int k = 0; k < D3; ++k) {
    const double v = useRe ? Tre[i][j][k] : Tim[i][j][k];
    nrm += v * v;
  }
  nrm = csqrt_(nrm);
  W3J<L1, L2, L3> R{};
  for (int i = 0; i < D1; ++i) for (int j = 0; j < D2; ++j) for (int k = 0; k < D3; ++k) {
    const double v = useRe ? Tre[i][j][k] : Tim[i][j][k];
    R.c[i][j][k] = (float)(scale * v / nrm);
  }
  return R;
}

// path coefficients sqrt((2*lo+1)/fan): fan0 = 64, fan1 = fan2 = 112
constexpr double K_C0 = 0.125;                 // sqrt(1/64)
constexpr double K_C1 = csqrt_(3.0 / 112.0);
constexpr double K_C2 = csqrt_(5.0 / 112.0);
constexpr auto C000s = make_w3j<0,0,0>(K_C0);
constexpr auto C101s = make_w3j<1,0,1>(K_C1);
constexpr auto C111s = make_w3j<1,1,1>(K_C1);
constexpr auto C121s = make_w3j<1,2,1>(K_C1);
constexpr auto C202s = make_w3j<2,0,2>(K_C2);
constexpr auto C212s = make_w3j<2,1,2>(K_C2);
constexpr auto C222s = make_w3j<2,2,2>(K_C2);

// ============================ device helpers ================================
// A fragment for V_WMMA_F32_16X16X4_F32:
//   lane L: M = u0 + (L&15); VGPR0 holds K = 4s+2h, VGPR1 holds K = 4s+2h+1 (h = L>>4)
template<int MV>
__device__ __forceinline__ void load_afrag(const float* __restrict__ W, int woff, int u0,
                                           int lane, v2f (&a)[MV/4]) {
  const int m = lane & 15;
  const int h = (lane >> 4) & 1;
#pragma unroll
  for (int s = 0; s < MV/4; ++s)
    a[s] = *reinterpret_cast<const v2f*>(W + woff + (u0 + m) * MV + 4*s + 2*h);
}

__device__ __forceinline__ v2f ldb(const float* __restrict__ base, int s, int dj) {
  v2f b;
  b.x = base[s * (64 * dj)];            // K = 4s+2h
  b.y = base[s * (64 * dj) + 16 * dj];  // K = 4s+2h+1
  return b;
}

// One 16(u) x 16(row) tile of yw[u,r] = sum_v w[u,v] * y[r, OOFF + v*DJ + j]
// ytt: this 16-row tile's LDS region, layout [c][r] with pitch 16 words.
// B-fragment loads software-pipelined 2 deep to cover LDS latency.
template<int MV, int DJ, int OOFF>
__device__ __forceinline__ v8f gemm_tile(const v2f (&a)[MV/4], const float* __restrict__ ytt,
                                         int lane, int j) {
  constexpr int NS = MV / 4;
  const int m = lane & 15;
  const int h = (lane >> 4) & 1;
  const float* base = ytt + (OOFF + 2*h*DJ + j) * 16 + m;
  v8f d = {0.f, 0.f, 0.f, 0.f, 0.f, 0.f, 0.f, 0.f};
  v2f bb[2];
  bb[0] = ldb(base, 0, DJ);
  if (NS > 1) bb[1] = ldb(base, 1, DJ);
#pragma unroll
  for (int s = 0; s < NS; ++s) {
    const v2f bcur = bb[s & 1];
    if (s + 2 < NS) bb[s & 1] = ldb(base, s + 2, DJ);
    d = WMMA_F32X4(a[s], bcur, d);
  }
  return d;
}

// Fused variants: one B stream feeds 2 / 3 independent accumulators (shared y0).
template<int MV, int DJ, int OOFF>
__device__ __forceinline__ void gemm_tile2(const v2f (&a0)[MV/4], const v2f (&a1)[MV/4],
                                           const float* __restrict__ ytt, int lane,
                                           v8f& d0, v8f& d1) {
  constexpr int NS = MV / 4;
  const int m = lane & 15;
  const int h = (lane >> 4) & 1;
  const float* base = ytt + (OOFF + 2*h*DJ) * 16 + m;     // j = 0
  v2f bb[2];
  bb[0] = ldb(base, 0, DJ);
  if (NS > 1) bb[1] = ldb(base, 1, DJ);
#pragma unroll
  for (int s = 0; s < NS; ++s) {
    const v2f bcur = bb[s & 1];
    if (s + 2 < NS) bb[s & 1] = ldb(base, s + 2, DJ);
    d0 = WMMA_F32X4(a0[s], bcur, d0);
    d1 = WMMA_F32X4(a1[s], bcur, d1);
  }
}
template<int MV, int DJ, int OOFF>
__device__ __forceinline__ void gemm_tile3(const v2f (&a0)[MV/4], const v2f (&a1)[MV/4],
                                           const v2f (&a2)[MV/4],
                                           const float* __restrict__ ytt, int lane,
                                           v8f& d0, v8f& d1, v8f& d2) {
  constexpr int NS = MV / 4;
  const int m = lane & 15;
  const int h = (lane >> 4) & 1;
  const float* base = ytt + (OOFF + 2*h*DJ) * 16 + m;     // j = 0
  v2f bb[2];
  bb[0] = ldb(base, 0, DJ);
  if (NS > 1) bb[1] = ldb(base, 1, DJ);
#pragma unroll
  for (int s = 0; s < NS; ++s) {
    const v2f bcur = bb[s & 1];
    if (s + 2 < NS) bb[s & 1] = ldb(base, s + 2, DJ);
    d0 = WMMA_F32X4(a0[s], bcur, d0);
    d1 = WMMA_F32X4(a1[s], bcur, d1);
    d2 = WMMA_F32X4(a2[s], bcur, d2);
  }
}

// O[g*DI+k] += sum_i Cscaled[i][j][k] * x[g*DI+i] * d[g]   (g = local u index)
template<int L1, int L2>
__device__ __forceinline__ void fold(const W3J<L1, L2, L1>& C, int j,
                                     const float* xr, const v8f& d, float* O) {
  constexpr int DI = 2*L1 + 1;
#pragma unroll
  for (int g = 0; g < 8; ++g) {
    const float dv = d[g];
#pragma unroll
    for (int k = 0; k < DI; ++k) {
      float acc = O[g*DI + k];
#pragma unroll
      for (int i = 0; i < DI; ++i) {
        const float cv = C.c[i][j][k];
        if (cv != 0.0f) acc = __builtin_fmaf(cv * xr[g*DI + i], dv, acc);
      }
      O[g*DI + k] = acc;
    }
  }
}

template<int NF>
__device__ __forceinline__ void load_x(const float* __restrict__ p, float (&x)[4*NF]) {
#pragma unroll
  for (int q = 0; q < NF; ++q) {
    const float4 v = reinterpret_cast<const float4*>(p)[q];
    x[4*q+0] = v.x; x[4*q+1] = v.y; x[4*q+2] = v.z; x[4*q+3] = v.w;
  }
}
template<int NF>
__device__ __forceinline__ void store_xo(float* __restrict__ p,
                                         const float (&x)[4*NF], const float (&o)[4*NF]) {
#pragma unroll
  for (int q = 0; q < NF; ++q) {
    float4 v;
    v.x = x[4*q+0] + o[4*q+0]; v.y = x[4*q+1] + o[4*q+1];
    v.z = x[4*q+2] + o[4*q+2]; v.w = x[4*q+3] + o[4*q+3];
    reinterpret_cast<float4*>(p)[q] = v;
  }
}

// ---- epilogues / jobs: one wave owns an (output irrep, 16-u tile) ----------
__device__ __forceinline__ void epi_i0(int u0, const v8f& d, int rbase, int n, int lane,
                                       const float* __restrict__ blk, float* __restrict__ out) {
  const int m = lane & 15, h = (lane >> 4) & 1;
  const int row = rbase + m;
  if (row < n) {
    const int xo = row * DB + u0 + 8*h;     // lane covers u = u0+8h+g, g=0..7
    float xr[8], O[8] = {};
    load_x<2>(blk + xo, xr);
    fold(C000s, 0, xr, d, O);
    store_xo<2>(out + xo, xr, O);
  }
}

__device__ __forceinline__ void job_i1_rest(int u0, const v8f& dP2, const v2f (&a3)[8],
                                            const v2f (&a4)[4], const float* __restrict__ ytt,
                                            int rbase, int n, int lane,
                                            const float* __restrict__ blk, float* __restrict__ out) {
  const int m = lane & 15, h = (lane >> 4) & 1;
  const int row = rbase + m;
  const bool act = row < n;
  const int xo = row * DB + 128 + (u0 + 8*h) * 3;
  float xr[24] = {}, O[24] = {};
  if (act) load_x<6>(blk + xo, xr);
  fold(C101s, 0, xr, dP2, O);
#pragma unroll
  for (int j = 0; j < 3; ++j) { v8f d = gemm_tile<32, 3,  64>(a3, ytt, lane, j); fold(C111s, j, xr, d, O); }
#pragma unroll
  for (int j = 0; j < 5; ++j) { v8f d = gemm_tile<16, 5, 160>(a4, ytt, lane, j); fold(C121s, j, xr, d, O); }
  if (act) store_xo<6>(out + xo, xr, O);
}

__device__ __forceinline__ void job_i2(int u0, const v2f (&a5)[16], const v2f (&a6)[8],
                                       const v2f (&a7)[4], const float* __restrict__ ytt,
                                       int rbase, int n, int lane,
                                       const float* __restrict__ blk, float* __restrict__ out) {
  const int m = lane & 15, h = (lane >> 4) & 1;
  const int row = rbase + m;
  const bool act = row < n;
  const int xo = row * DB + 320 + (u0 + 8*h) * 5;
  float xr[40] = {}, O[40] = {};
  if (act) load_x<10>(blk + xo, xr);
  { v8f d = gemm_tile<64, 1, 0>(a5, ytt, lane, 0); fold(C202s, 0, xr, d, O); }
#pragma unroll
  for (int j = 0; j < 3; ++j) { v8f d = gemm_tile<32, 3,  64>(a6, ytt, lane, j); fold(C212s, j, xr, d, O); }
#pragma unroll
  for (int j = 0; j < 5; ++j) { v8f d = gemm_tile<16, 5, 160>(a7, ytt, lane, j); fold(C222s, j, xr, d, O); }
  if (act) store_xo<10>(out + xo, xr, O);
}

// ---- stage 32 rows of `others` transposed into LDS + pass-through copy ----
// Full chunks use the CDNA5 async global->LDS path: each lane supplies its own
// LDS destination, so the DMA performs the [row][c] -> [c][row] transpose.
__device__ __forceinline__ void stage(const float* __restrict__ oth, float* __restrict__ outOth,
                                      float* __restrict__ yt, int r0, int n, int tid) {
  constexpr int ITERS = (CHUNK_ROWS * DO) / 256;   // 30
  if (r0 + CHUNK_ROWS <= n) {
    const uint32_t ybase = (uint32_t)(uintptr_t)yt;  // low 32 bits of a shared ptr = LDS byte offset
#pragma unroll
    for (int it = 0; it < ITERS; ++it) {
      const int idx = tid + it * 256;
      const int r32 = idx / DO;
      const int c   = idx - r32 * DO;
      const uint32_t lo = ybase + (uint32_t)(((((r32 >> 4) * DO + c) * 16) + (r32 & 15)) * 4);
      const float* g = oth + (size_t)(r0 + r32) * DO + c;
      asm volatile("global_load_async_to_lds_b32 %0, %1, off" :: "v"(lo), "v"(g) : "memory");
    }
    // pass-through copy of `others` (same lines -> cache hits), fully linear
#pragma unroll
    for (int it = 0; it < ITERS; ++it) {
      const size_t idx = (size_t)r0 * DO + tid + it * 256;
      outOth[idx] = oth[idx];
    }
#if __has_builtin(__builtin_amdgcn_s_wait_asynccnt)
    __builtin_amdgcn_s_wait_asynccnt(0);
#else
    asm volatile("s_wait_asynccnt 0x0" ::: "memory");
#endif
  } else {
    // tail chunk: scalar staging with zero fill
#pragma unroll
    for (int it = 0; it < ITERS; ++it) {
      const int idx = tid + it * 256;
      const int r32 = idx / DO;
      const int c   = idx - r32 * DO;
      const int row = r0 + r32;
      float v = 0.0f;
      if (row < n) {
        v = oth[(size_t)row * DO + c];
        outOth[(size_t)row * DO + c] = v;
      }
      yt[(((r32 >> 4) * DO + c) * 16) + (r32 & 15)] = v;
    }
  }
}

// =============================== kernel =====================================
__global__ void __launch_bounds__(256, 4)    // budget regs for 2 workgroups/WGP
tp_uvu_kernel(const float* __restrict__ blk, const float* __restrict__ oth,
              const float* __restrict__ W, float* __restrict__ out, int n) {
  __shared__ float yt[2 * TILE_WORDS];          // 30 KB, two 16-row tiles
  const int tid  = threadIdx.x;
  const int lane = tid & 31;
  const int wid  = tid >> 5;
  float* outOth = out + (size_t)n * DB;
  const int nchunks = (n + CHUNK_ROWS - 1) / CHUNK_ROWS;
  const v8f z8 = {0.f, 0.f, 0.f, 0.f, 0.f, 0.f, 0.f, 0.f};

  for (int ch = blockIdx.x; ch < nchunks; ch += gridDim.x) {
    const int r0 = ch * CHUNK_ROWS;
    stage(oth, outOth, yt, r0, n, tid);
    __syncthreads();

    if (wid < 2) {                              // 3 x scalar-irrep tiles (u0: 0..80), shared B
      v2f a0[16], a1[16], a2[16];
      const int ub = wid * 48;
      load_afrag<64>(W, 0, ub +  0, lane, a0);
      load_afrag<64>(W, 0, ub + 16, lane, a1);
      load_afrag<64>(W, 0, ub + 32, lane, a2);
#pragma unroll 1
      for (int t = 0; t < 2; ++t) {
        const float* ytt = yt + t * TILE_WORDS;
        const int rbase = r0 + 16 * t;
        v8f d0 = z8, d1 = z8, d2 = z8;
        gemm_tile3<64, 1, 0>(a0, a1, a2, ytt, lane, d0, d1, d2);
        SCHED_FENCE();
        epi_i0(ub +  0, d0, rbase, n, lane, blk, out);
        epi_i0(ub + 16, d1, rbase, n, lane, blk, out);
        epi_i0(ub + 32, d2, rbase, n, lane, blk, out);
        SCHED_FENCE();
      }
    } else if (wid < 4) {                       // scalar tile (u0: 96/112) + l=1 tile (u0: 0/16)
      v2f a0[16], a2[16], a3[8], a4[4];
      const int u0i0 = 96 + (wid - 2) * 16;
      const int u1   = (wid - 2) * 16;
      load_afrag<64>(W,     0, u0i0, lane, a0);
      load_afrag<64>(W,  8192, u1,   lane, a2);
      load_afrag<32>(W, 12288, u1,   lane, a3);
      load_afrag<16>(W, 14336, u1,   lane, a4);
#pragma unroll 1
      for (int t = 0; t < 2; ++t) {
        const float* ytt = yt + t * TILE_WORDS;
        const int rbase = r0 + 16 * t;
        v8f dA = z8, dP2 = z8;
        gemm_tile2<64, 1, 0>(a0, a2, ytt, lane, dA, dP2);   // p1 + p2 share the y0 B stream
        SCHED_FENCE();
        epi_i0(u0i0, dA, rbase, n, lane, blk, out);
        SCHED_FENCE();
        job_i1_rest(u1, dP2, a3, a4, ytt, rbase, n, lane, blk, out);
        SCHED_FENCE();
      }
    } else if (wid < 6) {                       // l=1 tile (u0: 32/48)
      v2f a2[16], a3[8], a4[4];
      const int u1 = (wid - 2) * 16;
      load_afrag<64>(W,  8192, u1, lane, a2);
      load_afrag<32>(W, 12288, u1, lane, a3);
      load_afrag<16>(W, 14336, u1, lane, a4);
#pragma unroll 1
      for (int t = 0; t < 2; ++t) {
        const float* ytt = yt + t * TILE_WORDS;
        const int rbase = r0 + 16 * t;
        v8f dP2 = gemm_tile<64, 1, 0>(a2, ytt, lane, 0);
        SCHED_FENCE();
        job_i1_rest(u1, dP2, a3, a4, ytt, rbase, n, lane, blk, out);
        SCHED_FENCE();
      }
    } else {                                    // l=2 tile (u0: 0/16)
      v2f a5[16], a6[8], a7[4];
      const int u2 = (wid - 6) * 16;
      load_afrag<64>(W, 15360, u2, lane, a5);
      load_afrag<32>(W, 17408, u2, lane, a6);
      load_afrag<16>(W, 18432, u2, lane, a7);
#pragma unroll 1
      for (int t = 0; t < 2; ++t) {
        const float* ytt = yt + t * TILE_WORDS;
        job_i2(u2, a5, a6, a7, ytt, r0 + 16*t, n, lane, blk, out);
        SCHED_FENCE();
      }
    }
    __syncthreads();
  }
}

// ============================== launcher ====================================
extern "C" void kernel_launch(void* const* d_in, const int* in_sizes, int n_in,
                              void* d_out, int out_size, void* d_ws, size_t ws_size,
                              hipStream_t stream) {
  (void)n_in; (void)out_size; (void)d_ws; (void)ws_size;
  const float* blk = (const float*)d_in[0];
  const float* oth = (const float*)d_in[1];
  const float* W   = (const float*)d_in[2];
  float* out = (float*)d_out;
  const int n = in_sizes[0] / DB;
  const int nchunks = (n + CHUNK_ROWS - 1) / CHUNK_ROWS;   // 3125 for n = 100000
  const int grid = nchunks < 625 ? nchunks : 625;          // 5 chunks per block
  tp_uvu_kernel<<<grid, 256, 0, stream>>>(blk, oth, W, out, n);
}